// ConcatCritic_3590592660028
// MI455X (gfx1250) — compile-verified
//
#include <hip/hip_runtime.h>

typedef float v2f __attribute__((ext_vector_type(2)));
typedef float v4f __attribute__((ext_vector_type(4)));
typedef float v8f __attribute__((ext_vector_type(8)));

#define BSZ 512   // batch (both i and j)
#define DX  128   // dim_x == dim_y
#define HID 512   // hidden

// ---------------------------------------------------------------------------
// Kernel 1: C[16x16 tile] = A(M=512,K=128) @ W(K=128,N=512) (+ bias), f32 WMMA.
// One wave (32 lanes) per 16x16 output tile; K-loop of 32 x V_WMMA_F32_16X16X4_F32.
//
// Fragment layouts per CDNA5 ISA 7.12.2 (32-bit, wave32):
//   A 16x4:  v0 = K=0 (lanes 0-15) / K=2 (lanes 16-31), v1 = K=1 / K=3; M = lane%16
//   B 4x16:  v0 = row K=0 (lanes 0-15) / K=2 (lanes 16-31), v1 = K=1 / K=3; N = lane%16
//   C/D 16x16: VGPR r -> M=r (lanes 0-15), M=r+8 (lanes 16-31); N = lane%16
// ---------------------------------------------------------------------------
__global__ __launch_bounds__(32)
void cc_gemm_wmma_f32(const float* __restrict__ A, const float* __restrict__ W,
                      const float* __restrict__ bias, float* __restrict__ C)
{
    const int lane = threadIdx.x;        // 0..31, EXEC all ones (WMMA requirement)
    const int tn   = blockIdx.x;         // 0..31 column tile (h)
    const int tm   = blockIdx.y;         // 0..31 row tile (batch)
    const int half = lane >> 4;          // 0 | 1
    const int l16  = lane & 15;

    v8f acc = {};
    #pragma unroll 4
    for (int k = 0; k < DX; k += 4) {
        // A fragment
        const float* ap = A + (tm * 16 + l16) * DX + k + half * 2;
        v2f a; a.x = ap[0]; a.y = ap[1];
        // B fragment (rows of W, column tn*16+l16)
        const float* bp = W + (k + half * 2) * HID + tn * 16 + l16;
        v2f b; b.x = bp[0]; b.y = bp[HID];
        acc = __builtin_amdgcn_wmma_f32_16x16x4_f32(
                  /*neg_a=*/false, a, /*neg_b=*/false, b,
                  /*c_mod=*/(short)0, acc, /*reuse_a=*/false, /*reuse_b=*/false);
    }

    const int col = tn * 16 + l16;
    const float bv = bias ? bias[col] : 0.0f;
    #pragma unroll
    for (int r = 0; r < 8; ++r) {
        const int row = tm * 16 + r + half * 8;
        C[row * HID + col] = acc[r] + bv;
    }
}

// ---------------------------------------------------------------------------
// Kernel 2: scores[i,j] = sum_h relu(hx[i,h] + hy[j,h]) * w2[h] + b2.
// 256 threads = one 16x16 (i,j) tile; h chunked by 256 through LDS.
// LDS row stride 260 floats (260 % 64 == 4 banks, 16B aligned) so the 16
// distinct hy rows read per wave hit disjoint bank groups with ds_load_b128.
// ---------------------------------------------------------------------------
__global__ __launch_bounds__(256)
void cc_pair_scores(const float* __restrict__ hx, const float* __restrict__ hy,
                    const float* __restrict__ W2, const float* __restrict__ b2,
                    float* __restrict__ out)
{
    constexpr int CH = 256;       // h-chunk length
    constexpr int ST = CH + 4;    // LDS row stride (floats)
    __shared__ float sx[16 * ST];
    __shared__ float sy[16 * ST];
    __shared__ float sw[CH];

    const int tid = threadIdx.x;
    const int i0  = blockIdx.y * 16;
    const int j0  = blockIdx.x * 16;
    const int tj  = tid & 15;
    const int ti  = tid >> 4;

    float acc = 0.0f;
    for (int hb = 0; hb < HID; hb += CH) {
        // Stage 16 rows x 256 floats of hx and hy (1024 float4 each / 256 thr)
        #pragma unroll
        for (int v = tid; v < 16 * (CH / 4); v += 256) {
            const int row = v >> 6;          // / (CH/4)
            const int c4  = v & 63;          // % (CH/4)
            *(v4f*)(sx + row * ST + c4 * 4) =
                *(const v4f*)(hx + (i0 + row) * HID + hb + c4 * 4);
            *(v4f*)(sy + row * ST + c4 * 4) =
                *(const v4f*)(hy + (j0 + row) * HID + hb + c4 * 4);
        }
        if (tid < CH / 4)
            *(v4f*)(sw + tid * 4) = *(const v4f*)(W2 + hb + tid * 4);
        __syncthreads();

        const float* rx = sx + ti * ST;
        const float* ry = sy + tj * ST;
        #pragma unroll 4
        for (int h = 0; h < CH; h += 4) {
            const v4f xv = *(const v4f*)(rx + h);
            const v4f yv = *(const v4f*)(ry + h);
            const v4f wv = *(const v4f*)(sw + h);
            #pragma unroll
            for (int u = 0; u < 4; ++u) {
                float s = xv[u] + yv[u];
                s = fmaxf(s, 0.0f);
                acc = fmaf(s, wv[u], acc);
            }
        }
        __syncthreads();
    }

    out[(i0 + ti) * BSZ + (j0 + tj)] = acc + b2[0];
}

// ---------------------------------------------------------------------------
// Launch: GEMM hx (bias folded) + GEMM hy into workspace, then pair scores.
// Workspace use: 2 x 512 x 512 f32 = 2 MB.
// ---------------------------------------------------------------------------
extern "C" void kernel_launch(void* const* d_in, const int* in_sizes, int n_in,
                              void* d_out, int out_size, void* d_ws, size_t ws_size,
                              hipStream_t stream) {
    const float* x  = (const float*)d_in[0];   // [512,128]
    const float* y  = (const float*)d_in[1];   // [512,128]
    const float* W1 = (const float*)d_in[2];   // [256,512]
    const float* b1 = (const float*)d_in[3];   // [512]
    const float* W2 = (const float*)d_in[4];   // [512,1]
    const float* b2 = (const float*)d_in[5];   // [1]
    float* out = (float*)d_out;                // [512,512]

    float* hx = (float*)d_ws;                  // 512*512 f32
    float* hy = hx + BSZ * HID;                // 512*512 f32

    dim3 g1(HID / 16, BSZ / 16);               // 32 x 32 tiles, 1 wave each
    cc_gemm_wmma_f32<<<g1, 32, 0, stream>>>(x, W1, b1, hx);
    cc_gemm_wmma_f32<<<g1, 32, 0, stream>>>(y, W1 + DX * HID, nullptr, hy);

    dim3 g2(BSZ / 16, BSZ / 16);               // 32 x 32 output tiles
    cc_pair_scores<<<g2, 256, 0, stream>>>(hx, hy, W2, b2, out);
}